// NLM_29918742184056
// MI455X (gfx1250) — compile-verified
//
#include <hip/hip_runtime.h>
#include <hip/hip_bf16.h>

// ---------------------------------------------------------------------------
// NLM on gfx1250: one wave32 per 16x16 output tile.
// 7x7 box-mean expressed as two banded-matrix matmuls on V_WMMA_F32_16X16X4_F32.
// All intermediates wave-private in LDS; weight scale -log2(e)/(49*H^2) folded
// into the horizontal band matrix so w = exp2(D) (single v_exp_f32 per value).
// ---------------------------------------------------------------------------

typedef __attribute__((ext_vector_type(2))) float v2f;
typedef __attribute__((ext_vector_type(8))) float v8f;

#define IMG  1024
#define TPW  7          // template window
#define THW  3          // template half
#define SW   21         // search window
#define SHW  10         // search half
#define NOFF (SW * SW)  // 441
#define HALO 22         // 16 + 2*THW
#define KPAD 24         // K padded to multiple of 4

// reflect-101 (jnp "reflect") index map; |excursion| <= 13 so one pass suffices
__device__ __forceinline__ int refl(int i) {
    i = (i < 0) ? -i : i;
    return (i >= IMG) ? (2 * IMG - 2 - i) : i;
}
__device__ __forceinline__ float clamp01(float v) {
    return fminf(fmaxf(v, 0.0f), 1.0f);
}

__global__ __launch_bounds__(128) void nlm_wmma_kernel(const float* __restrict__ x,
                                                       float* __restrict__ out) {
    // per-wave LDS slabs (4 waves / block)
    __shared__ float s_cen[4][HALO * KPAD];  // clipped center halo   22x22 (stride 24)
    __shared__ float s_sq [4][KPAD * 32];    // squared diff          24x32, K-pad rows/cols = 0
    __shared__ float s_v  [4][16 * 32];      // vertical box sums V   16x24 valid (stride 32)
    __shared__ float s_sh [4][16 * 16];      // shifted center values 16x16

    const int wave = threadIdx.x >> 5;
    const int lane = threadIdx.x & 31;
    const int gwave = blockIdx.x * 4 + wave;
    const int img  = gwave >> 12;            // 4096 tiles per image
    const int tile = gwave & 4095;
    const int r0 = (tile >> 6) << 4;
    const int c0 = (tile & 63) << 4;

    const float* __restrict__ src = x   + (size_t)img * IMG * IMG;
    float*       __restrict__ dst = out + (size_t)img * IMG * IMG;

    float* cen = s_cen[wave];
    float* sq  = s_sq [wave];
    float* vb  = s_v  [wave];
    float* shc = s_sh [wave];

    // zero sq buffer once: rows 22..23 and cols 22..31 stay 0 forever (K padding)
    for (int t = lane; t < KPAD * 32; t += 32) sq[t] = 0.0f;

    // center halo (offset-invariant): cen[i][j] = clip(x[refl(r0-3+i)][refl(c0-3+j)])
    for (int t = lane; t < HALO * HALO; t += 32) {
        int i = t / HALO, j = t - i * HALO;
        int ym = refl(r0 - THW + i), xm = refl(c0 - THW + j);
        cen[i * KPAD + j] = clamp01(src[ym * IMG + xm]);
    }

    // per-lane WMMA coordinates
    const int m     = lane & 15;             // A-row M / B-col N / D-col N
    const int khalf = lane >> 4;             // K sub-offset (+2) and D row (+8)

    // band value for horizontal pass: folds 1/49, -1/H^2 and log2(e) -> w = exp2(D)
    const float HP = 7.0f / 255.0f;
    const float WBAND = -1.4426950408889634f / (49.0f * HP * HP);

    v8f wsum = {0.f, 0.f, 0.f, 0.f, 0.f, 0.f, 0.f, 0.f};
    v8f vsum = {0.f, 0.f, 0.f, 0.f, 0.f, 0.f, 0.f, 0.f};

    for (int off = 0; off < NOFF; ++off) {
        const int dy = off / SW - SHW;
        const int dx = off - (off / SW) * SW - SHW;

        // build shifted halo -> sq (and stash the 16x16 shifted center for accumulation)
        for (int t = lane; t < HALO * HALO; t += 32) {
            int i = t / HALO, j = t - i * HALO;
            int ym = refl(r0 - THW + i), xm = refl(c0 - THW + j);   // box-filter reflect
            int ys = refl(ym + dy),      xs = refl(xm + dx);        // search-pad reflect
            float sv = clamp01(src[ys * IMG + xs]);
            float d  = cen[i * KPAD + j] - sv;
            sq[i * 32 + j] = d * d;
            if ((unsigned)(i - THW) < 16u && (unsigned)(j - THW) < 16u)
                shc[(i - THW) * 16 + (j - THW)] = sv;
        }

        // ---- vertical pass: V = Mv(16x24) x sq(24x22), two 16-col halves -------
        #pragma unroll
        for (int half = 0; half < 2; ++half) {
            v8f acc = {0.f, 0.f, 0.f, 0.f, 0.f, 0.f, 0.f, 0.f};
            #pragma unroll
            for (int k0 = 0; k0 < 6; ++k0) {
                int kk = 4 * k0 + 2 * khalf;
                // A = Mv chunk (constant band): A[m][kk..kk+1]
                v2f a;
                a.x = ((unsigned)(kk     - m) <= 6u) ? 1.0f : 0.0f;
                a.y = ((unsigned)(kk + 1 - m) <= 6u) ? 1.0f : 0.0f;
                // B = sq chunk from LDS: B[kk..kk+1][m] (col half offset)
                v2f b;
                b.x = sq[kk * 32       + 16 * half + m];
                b.y = sq[(kk + 1) * 32 + 16 * half + m];
                acc = __builtin_amdgcn_wmma_f32_16x16x4_f32(
                          false, a, false, b, (short)0, acc, false, false);
            }
            // D(C/D layout) -> V buffer rows g+8*khalf, cols m+16*half
            #pragma unroll
            for (int g = 0; g < 8; ++g)
                vb[(g + 8 * khalf) * 32 + 16 * half + m] = acc[g];
        }

        // ---- horizontal pass: D = V(16x24) x Mh(24x16); Mh band = WBAND --------
        v8f acc2 = {0.f, 0.f, 0.f, 0.f, 0.f, 0.f, 0.f, 0.f};
        #pragma unroll
        for (int k0 = 0; k0 < 6; ++k0) {
            int kk = 4 * k0 + 2 * khalf;
            v2f a;                                   // A = V: A[m][kk..kk+1]
            a.x = vb[m * 32 + kk];
            a.y = vb[m * 32 + kk + 1];
            v2f b;                                   // B = Mh band: B[kk..kk+1][m]
            b.x = ((unsigned)(kk     - m) <= 6u) ? WBAND : 0.0f;
            b.y = ((unsigned)(kk + 1 - m) <= 6u) ? WBAND : 0.0f;
            acc2 = __builtin_amdgcn_wmma_f32_16x16x4_f32(
                       false, a, false, b, (short)0, acc2, false, false);
        }

        // ---- w = exp2(D) ; accumulate in C/D layout -----------------------------
        #pragma unroll
        for (int g = 0; g < 8; ++g) {
            float w = exp2f(acc2[g]);                // D already = -dist/H^2 * log2e
            wsum[g] += w;
            vsum[g] += w * shc[(g + 8 * khalf) * 16 + m];
        }
    }

    // final: out = clip(vsum/wsum), D layout -> row-major global
    #pragma unroll
    for (int g = 0; g < 8; ++g) {
        int r = r0 + g + 8 * khalf;
        int c = c0 + m;
        dst[r * IMG + c] = clamp01(vsum[g] / wsum[g]);
    }
}

extern "C" void kernel_launch(void* const* d_in, const int* in_sizes, int n_in,
                              void* d_out, int out_size, void* d_ws, size_t ws_size,
                              hipStream_t stream) {
    (void)in_sizes; (void)n_in; (void)d_ws; (void)ws_size; (void)out_size;
    const float* x = (const float*)d_in[0];
    float* out = (float*)d_out;
    // 4 images * 64*64 tiles = 16384 waves; 4 waves (128 threads) per block
    nlm_wmma_kernel<<<4096, 128, 0, stream>>>(x, out);
}